// ExternalLSTM_87737591922706
// MI455X (gfx1250) — compile-verified
//
#include <hip/hip_runtime.h>
#include <hip/hip_bf16.h>

// ---------------------------------------------------------------------------
// Types
// ---------------------------------------------------------------------------
typedef __attribute__((ext_vector_type(2))) float f32x2;
typedef __attribute__((ext_vector_type(4))) float f32x4;
typedef __attribute__((ext_vector_type(8))) float f32x8;

#if defined(__has_builtin)
#if __has_builtin(__builtin_amdgcn_wmma_f32_16x16x4_f32)
#define HAS_WMMA_F32X4 1
#endif
#endif

#define IN_F 32
#define NUM_NODES 100000
#define OUT_F 8
#define HDIM 64
#define TSTEPS 4096
#define G4 256                     // 4*HDIM
#define DPROJ (NUM_NODES * OUT_F)  // 800000

// Fast activations: v_exp_f32 + v_rcp_f32 only (serial critical path in scan).
static __device__ __forceinline__ float fast_sigmoid(float x) {
    return __builtin_amdgcn_rcpf(1.0f + __expf(-x));
}
static __device__ __forceinline__ float fast_tanh(float x) {
    const float e2 = __expf(2.0f * x);
    return 1.0f - 2.0f * __builtin_amdgcn_rcpf(e2 + 1.0f);
}
static __device__ __forceinline__ f32x2 fma2(f32x2 a, f32x2 b, f32x2 c) {
    return __builtin_elementwise_fma(a, b, c);
}

// ---------------------------------------------------------------------------
// Kernel 1: gx0[t,g] = sum_k x[t,k] * w_ih0[g,k] + b_ih0[g] + b_hh0[g]
// GEMM 4096x256x32 (f32). One wave32 per 16x16 tile, 8x V_WMMA_F32_16X16X4_F32.
//   A (16x4 f32):  lanes 0-15: M=lane, VGPR0->K=0, VGPR1->K=1
//                  lanes 16-31: M=lane-16, VGPR0->K=2, VGPR1->K=3
//   B (4x16 f32):  mirrored; C/D: VGPR v, lanes 0-15: M=v, lanes 16-31: M=v+8
// ---------------------------------------------------------------------------
__global__ __launch_bounds__(128)
void gx0_wmma_kernel(const float* __restrict__ x,
                     const float* __restrict__ w_ih0,
                     const float* __restrict__ b_ih0,
                     const float* __restrict__ b_hh0,
                     float* __restrict__ gx0) {
    const int wave = (blockIdx.x * blockDim.x + threadIdx.x) >> 5;
    const int lane = threadIdx.x & 31;
    const int tm = wave >> 4;            // 256 tiles along T
    const int tn = wave & 15;            // 16 tiles along 4H
    if (tm >= TSTEPS / 16) return;
    const int m0 = tm * 16;
    const int n0 = tn * 16;
    const int half = lane >> 4;          // 0 | 1
    const int l15  = lane & 15;

#if defined(HAS_WMMA_F32X4)
    f32x8 acc = {};
#pragma unroll
    for (int kc = 0; kc < 8; ++kc) {
        const int ka = kc * 4 + half * 2;
        f32x2 a, b;
        a.x = x[(m0 + l15) * IN_F + ka];
        a.y = x[(m0 + l15) * IN_F + ka + 1];
        b.x = w_ih0[(n0 + l15) * IN_F + ka];
        b.y = w_ih0[(n0 + l15) * IN_F + ka + 1];
        acc = __builtin_amdgcn_wmma_f32_16x16x4_f32(
            /*neg_a=*/false, a, /*neg_b=*/false, b,
            /*c_mod=*/(short)0, acc, /*reuse_a=*/false, /*reuse_b=*/false);
    }
#pragma unroll
    for (int v = 0; v < 8; ++v) {
        const int row = v + half * 8;
        const int n = n0 + l15;
        gx0[(m0 + row) * G4 + n] = acc[v] + b_ih0[n] + b_hh0[n];
    }
#else
#pragma unroll
    for (int v = 0; v < 8; ++v) {
        const int row = v + half * 8;
        const int n = n0 + l15;
        float acc = 0.0f;
        for (int k = 0; k < IN_F; ++k)
            acc = fmaf(x[(m0 + row) * IN_F + k], w_ih0[n * IN_F + k], acc);
        gx0[(m0 + row) * G4 + n] = acc + b_ih0[n] + b_hh0[n];
    }
#endif
}

// ---------------------------------------------------------------------------
// Kernel 2: fused 2-layer LSTM scan, single workgroup (512 threads = 16 waves).
// Pipeline: iteration `it` -> layer0 step `it`, layer1 step `it-1` (consumes
// the current h0 ping-pong buffer == ys0[it-1]).
//   thread g       (g<256): gate0[g] = gx0[it,g] + w_hh0[g]·h0   (64 MACs)
//                           part1[g] = w_ih1[g][0:32]·h0[0:32]   (32 MACs)
//   thread 256+g          : part2[g] = w_ih1[g][32:64]·h0[32:64]
//                                      + w_hh1[g]·h1 + b1[g]     (96 MACs)
// Weights in VGPRs (96 f32/lane), h via b128 LDS broadcast, packed f32x2 FMAs.
// gx0 is register double-buffered one iteration ahead (L2 latency off the
// serial path) and prefetched into WGP$ 8 iterations ahead.
// ---------------------------------------------------------------------------
__global__ __launch_bounds__(512, 1)
void lstm_scan_kernel(const float* __restrict__ gx0,
                      const float* __restrict__ w_hh0,
                      const float* __restrict__ w_ih1,
                      const float* __restrict__ w_hh1,
                      const float* __restrict__ b_ih1,
                      const float* __restrict__ b_hh1,
                      float* __restrict__ h1_out,
                      float* __restrict__ c1_out) {
    __shared__ __align__(16) float h0buf[2 * HDIM];  // ping-pong ys0
    __shared__ __align__(16) float c0s[HDIM];
    __shared__ __align__(16) float h1s[HDIM];
    __shared__ __align__(16) float c1s[HDIM];
    __shared__ __align__(16) float gate0[G4];
    __shared__ __align__(16) float part1[G4];
    __shared__ __align__(16) float part2[G4];
    __shared__ __align__(16) float bs1[G4];

    const int tid = threadIdx.x;
    const int g = tid & 255;
    const bool isL0 = tid < 256;

    if (tid < 2 * HDIM) h0buf[tid] = 0.0f;
    if (tid < HDIM) { c0s[tid] = 0.0f; h1s[tid] = 0.0f; c1s[tid] = 0.0f; }
    if (tid < G4) bs1[tid] = b_ih1[tid] + b_hh1[tid];

    // Weight preload into registers.
    f32x2 wb[32];  // "big" row: w_hh0[g] (L0) or w_hh1[g] (L1)
    f32x2 ws[16];  // "small" half-row of w_ih1[g]
    const float* bigsrc = isL0 ? (w_hh0 + g * HDIM) : (w_hh1 + g * HDIM);
    const float* smsrc  = isL0 ? (w_ih1 + g * HDIM) : (w_ih1 + g * HDIM + 32);
#pragma unroll
    for (int k = 0; k < 32; ++k) { wb[k].x = bigsrc[2 * k]; wb[k].y = bigsrc[2 * k + 1]; }
#pragma unroll
    for (int k = 0; k < 16; ++k) { ws[k].x = smsrc[2 * k]; ws[k].y = smsrc[2 * k + 1]; }

    float gxv = isL0 ? gx0[g] : 0.0f;   // gx0 for iteration 0
    __syncthreads();

    int cur = 0;
    for (int it = 0; it <= TSTEPS; ++it) {
        // Issue next iteration's gx0 load now; consume after the next barrier.
        float gxn = 0.0f;
        if (isL0) {
            if (it + 1 < TSTEPS) gxn = gx0[(it + 1) * G4 + g];
            const int pft = (it + 8 < TSTEPS) ? (it + 8) : (TSTEPS - 1);
            __builtin_prefetch(gx0 + (size_t)pft * G4 + g, 0, 3);
        }

        const f32x4* h0c4 = (const f32x4*)(h0buf + cur * HDIM);
        const f32x4* h1s4 = (const f32x4*)h1s;

        // -------- gate phase --------
        if (isL0) {
            f32x2 a2 = {0.0f, 0.0f};
            f32x2 p2 = {0.0f, 0.0f};
#pragma unroll
            for (int k = 0; k < 16; ++k) {
                const f32x4 h4 = h0c4[k];
                a2 = fma2(wb[2 * k],     h4.xy, a2);
                a2 = fma2(wb[2 * k + 1], h4.zw, a2);
                if (k < 8) {
                    p2 = fma2(ws[2 * k],     h4.xy, p2);
                    p2 = fma2(ws[2 * k + 1], h4.zw, p2);
                }
            }
            part1[g] = p2.x + p2.y;
            if (it < TSTEPS) gate0[g] = gxv + a2.x + a2.y;
        } else {
            f32x2 a2 = {0.0f, 0.0f};
#pragma unroll
            for (int k = 0; k < 16; ++k) {
                const f32x4 h4 = h1s4[k];
                a2 = fma2(wb[2 * k],     h4.xy, a2);
                a2 = fma2(wb[2 * k + 1], h4.zw, a2);
            }
#pragma unroll
            for (int k = 0; k < 8; ++k) {
                const f32x4 h4 = h0c4[8 + k];
                a2 = fma2(ws[2 * k],     h4.xy, a2);
                a2 = fma2(ws[2 * k + 1], h4.zw, a2);
            }
            part2[g] = bs1[g] + a2.x + a2.y;
        }
        __syncthreads();

        // -------- update phase --------
        const int nxt = cur ^ 1;
        if (tid < HDIM) {
            if (it < TSTEPS) {
                const int j = tid;
                const float i_ = fast_sigmoid(gate0[j]);
                const float f_ = fast_sigmoid(gate0[HDIM + j]);
                const float gg = fast_tanh(gate0[2 * HDIM + j]);
                const float o_ = fast_sigmoid(gate0[3 * HDIM + j]);
                const float c = f_ * c0s[j] + i_ * gg;
                c0s[j] = c;
                h0buf[nxt * HDIM + j] = o_ * fast_tanh(c);
            }
        } else if (tid < 2 * HDIM) {
            if (it >= 1) {
                const int j = tid - HDIM;
                const float i_ = fast_sigmoid(part1[j] + part2[j]);
                const float f_ = fast_sigmoid(part1[HDIM + j] + part2[HDIM + j]);
                const float gg = fast_tanh(part1[2 * HDIM + j] + part2[2 * HDIM + j]);
                const float o_ = fast_sigmoid(part1[3 * HDIM + j] + part2[3 * HDIM + j]);
                const float c = f_ * c1s[j] + i_ * gg;
                c1s[j] = c;
                h1s[j] = o_ * fast_tanh(c);
            }
        }
        __syncthreads();
        cur = nxt;
        gxv = gxn;
    }

    if (tid < HDIM) { h1_out[tid] = h1s[tid]; c1_out[tid] = c1s[tid]; }
}

// ---------------------------------------------------------------------------
// Kernel 3/4: out[d] = b[d] + vec · W[d,:]   (D = 800000, HBM-bound: 205 MB
// per matrix). 16 lanes per row -> every b128 load instruction covers a fully
// contiguous 512 B (2 rows/wave): transaction-minimal streaming at 23.3 TB/s.
// Non-temporal loads keep the one-shot weight stream from thrashing caches.
// ---------------------------------------------------------------------------
__global__ __launch_bounds__(256)
void proj_kernel(const float* __restrict__ vec,   // [64]
                 const float* __restrict__ W,     // [D, 64]
                 const float* __restrict__ bias,  // [D]
                 float* __restrict__ out, int D) {
    __shared__ float v[HDIM];
    const int tid = threadIdx.x;
    if (tid < HDIM) v[tid] = vec[tid];
    __syncthreads();

    const int r = (blockIdx.x * blockDim.x + tid) >> 4;  // row (16 lanes/row)
    const int q = tid & 15;                              // lane within row
    if (r >= D) return;

    const f32x4 w = __builtin_nontemporal_load(
        (const f32x4*)(W + (size_t)r * HDIM + q * 4));
    float part = w.x * v[4 * q + 0];
    part = fmaf(w.y, v[4 * q + 1], part);
    part = fmaf(w.z, v[4 * q + 2], part);
    part = fmaf(w.w, v[4 * q + 3], part);

    // Reduce across the 16 lanes of this row.
    part += __shfl_xor(part, 1, 16);
    part += __shfl_xor(part, 2, 16);
    part += __shfl_xor(part, 4, 16);
    part += __shfl_xor(part, 8, 16);

    if (q == 0) out[r] = part + bias[r];
}

// ---------------------------------------------------------------------------
// Launch
// ---------------------------------------------------------------------------
extern "C" void kernel_launch(void* const* d_in, const int* in_sizes, int n_in,
                              void* d_out, int out_size, void* d_ws, size_t ws_size,
                              hipStream_t stream) {
    const float* x     = (const float*)d_in[0];
    const float* w_ih0 = (const float*)d_in[1];
    const float* w_hh0 = (const float*)d_in[2];
    const float* b_ih0 = (const float*)d_in[3];
    const float* b_hh0 = (const float*)d_in[4];
    const float* w_ih1 = (const float*)d_in[5];
    const float* w_hh1 = (const float*)d_in[6];
    const float* b_ih1 = (const float*)d_in[7];
    const float* b_hh1 = (const float*)d_in[8];
    const float* w_eh  = (const float*)d_in[9];
    const float* b_eh  = (const float*)d_in[10];
    const float* w_es  = (const float*)d_in[11];
    const float* b_es  = (const float*)d_in[12];

    float* ws  = (float*)d_ws;
    float* gx0 = ws;                        // 4096*256 floats = 4 MB
    float* h1w = ws + (size_t)TSTEPS * G4;  // 64 floats
    float* c1w = h1w + HDIM;                // 64 floats
    float* out = (float*)d_out;

    // Phase A: input-gate GEMM via WMMA (4096 wave-tiles, 4 waves/block).
    gx0_wmma_kernel<<<(TSTEPS / 16) * 16 / 4, 128, 0, stream>>>(
        x, w_ih0, b_ih0, b_hh0, gx0);

    // Phase B: fused sequential 2-layer scan (single workgroup).
    lstm_scan_kernel<<<1, 512, 0, stream>>>(
        gx0, w_hh0, w_ih1, w_hh1, b_ih1, b_hh1, h1w, c1w);

    // Phase C: streaming GEMVs (16 lanes/row). Output order: (emb_state, emb_hidden).
    const int threadsPerMat = DPROJ * 16;   // 12.8M threads
    proj_kernel<<<threadsPerMat / 256, 256, 0, stream>>>(c1w, w_es, b_es, out, DPROJ);
    proj_kernel<<<threadsPerMat / 256, 256, 0, stream>>>(h1w, w_eh, b_eh, out + DPROJ, DPROJ);
}